// GATGRUCell_58205396795990
// MI455X (gfx1250) — compile-verified
//
#include <hip/hip_runtime.h>
#include <hip/hip_bf16.h>
#include <cstdint>

// ---------------- problem constants ----------------
#define BB 4
#define NN 2000
#define EE 64000
#define HH 64
#define HX 192          // 3*H
#define BNH 512000      // B*N*H
#define BN  8000        // B*N
#define SCALE_F 0.125f  // 64^-0.5
#define NEG_SLOPE_F 0.2f

typedef __attribute__((ext_vector_type(2))) float v2f;
typedef __attribute__((ext_vector_type(8))) float v8f;
typedef __attribute__((ext_vector_type(4))) unsigned v4u;
typedef __attribute__((ext_vector_type(4))) int v4i;
typedef __attribute__((ext_vector_type(8))) int v8i;

__device__ __forceinline__ v8f wmma4(v2f a, v2f b, v8f c) {
  // D = A(16x4 f32) * B(4x16 f32) + C(16x16 f32)
  return __builtin_amdgcn_wmma_f32_16x16x4_f32(false, a, false, b, (short)0, c,
                                               false, false);
}

__device__ __forceinline__ float rowmax16(float v) {
#pragma unroll
  for (int m = 1; m < 16; m <<= 1) v = fmaxf(v, __shfl_xor(v, m, 32));
  return v;
}
__device__ __forceinline__ float rowsum16(float v) {
#pragma unroll
  for (int m = 1; m < 16; m <<= 1) v += __shfl_xor(v, m, 32);
  return v;
}

// monotonic float<->uint encoding for atomic max over signed floats
__device__ __forceinline__ unsigned fenc(float f) {
  unsigned u = __float_as_uint(f);
  return (u & 0x80000000u) ? ~u : (u | 0x80000000u);
}
__device__ __forceinline__ float fdec(unsigned e) {
  unsigned u = (e & 0x80000000u) ? (e ^ 0x80000000u) : ~e;
  return __uint_as_float(u);
}
#define ENC_NEG_INF 0x007FFFFFu  // fenc(-inf)

// TDM: DMA one 16x64 f32 row-major tile at `gptr` into LDS offset `lds_byte`,
// padding 4 DWORDs after every 64 DWORDs (=> LDS row stride 68 floats, no
// 64-bank conflicts on the stride-4 B-fragment gathers).
__device__ __forceinline__ void tdm_load_tile16x64(unsigned lds_byte,
                                                   const float* gptr) {
  unsigned long long ga = (unsigned long long)(uintptr_t)gptr;
  v4u g0;
  g0[0] = 1u;                                    // count=1, user mode, no gather
  g0[1] = lds_byte;                              // D#.lds_addr
  g0[2] = (unsigned)(ga & 0xFFFFFFFFu);          // global_addr[31:0]
  g0[3] = (unsigned)((ga >> 32) & 0x01FFFFFFu)   // global_addr[56:32]
          | (2u << 30);                          // type = 2 ("image")
  v8i g1;
  g1[0] = (int)((2u << 16)      // data_size = 4 bytes
                | (1u << 20)    // pad_enable
                | (5u << 22)    // pad_interval: 64 DWORDs
                | (3u << 25));  // pad_amount: 4 DWORDs
  g1[1] = (int)(64u << 16);     // tensor_dim0 = 64 (bits 79:48, low half)
  g1[2] = (int)(16u << 16);     // tensor_dim1 = 16 (bits 111:80, low half)
  g1[3] = (int)(64u << 16);     // tile_dim0 = 64 (bits 127:112)
  g1[4] = 16;                   // tile_dim1 = 16 (bits 143:128); tile_dim2 = 0
  g1[5] = 64;                   // tensor_dim0_stride = 64 (bits 207:160)
  g1[6] = 0;
  g1[7] = 0;
  v4i z = {0, 0, 0, 0};
#if __clang_major__ >= 23
  v8i z8 = {0, 0, 0, 0, 0, 0, 0, 0};
  __builtin_amdgcn_tensor_load_to_lds(g0, g1, z, z, z8, 0);
#else
  __builtin_amdgcn_tensor_load_to_lds(g0, g1, z, z, 0);
#endif
}

// ---------------- 1) q/k/v projections (K=2 or 4: scalar) ----------------
__global__ void proj_kernel(
    const float* __restrict__ in, const float* Wqa, const float* bqa,
    const float* Wka, const float* bka, const float* Wva, const float* bva,
    const float* Wqb, const float* bqb, const float* Wkb, const float* bkb,
    const float* Wvb, const float* bvb, const float* Wqc, const float* bqc,
    const float* Wkc, const float* bkc, const float* Wvc, const float* bvc,
    float* __restrict__ qkv) {
  int i = blockIdx.x * 256 + threadIdx.x;
  if (i >= BNH) return;
  int h = i & 63;
  size_t row = (size_t)(i >> 6);  // b*N+n
  const float* r = in + row * 6;
  float m0 = r[0], m1 = r[1], l0 = r[2], l1 = r[3], t0 = r[4], t1 = r[5];
  // attn a: Q = metrics, K/V = [logs, traces]
  qkv[0 * BNH + i] = m0 * Wqa[h] + m1 * Wqa[64 + h] + bqa[h];
  qkv[1 * BNH + i] = l0 * Wka[h] + l1 * Wka[64 + h] + t0 * Wka[128 + h] +
                     t1 * Wka[192 + h] + bka[h];
  qkv[2 * BNH + i] = l0 * Wva[h] + l1 * Wva[64 + h] + t0 * Wva[128 + h] +
                     t1 * Wva[192 + h] + bva[h];
  // attn b: Q = logs, K/V = [metrics, traces]
  qkv[3 * BNH + i] = l0 * Wqb[h] + l1 * Wqb[64 + h] + bqb[h];
  qkv[4 * BNH + i] = m0 * Wkb[h] + m1 * Wkb[64 + h] + t0 * Wkb[128 + h] +
                     t1 * Wkb[192 + h] + bkb[h];
  qkv[5 * BNH + i] = m0 * Wvb[h] + m1 * Wvb[64 + h] + t0 * Wvb[128 + h] +
                     t1 * Wvb[192 + h] + bvb[h];
  // attn c: Q = traces, K/V = [metrics, logs]
  qkv[6 * BNH + i] = t0 * Wqc[h] + t1 * Wqc[64 + h] + bqc[h];
  qkv[7 * BNH + i] = m0 * Wkc[h] + m1 * Wkc[64 + h] + l0 * Wkc[128 + h] +
                     l1 * Wkc[192 + h] + bkc[h];
  qkv[8 * BNH + i] = m0 * Wvc[h] + m1 * Wvc[64 + h] + l0 * Wvc[128 + h] +
                     l1 * Wvc[192 + h] + bvc[h];
}

// ---------------- 2) flash attention, 1 wave per 16-query tile ----------------
// K tiles arrive via TDM (tensor_load_to_lds, double-buffered, TENSORcnt).
__global__ __launch_bounds__(32) void attn_kernel(const float* __restrict__ qkv,
                                                  float* __restrict__ abc) {
  const int tile = blockIdx.x;  // 0..124
  const int b = blockIdx.y;     // 0..3
  const int at = blockIdx.z;    // 0..2
  const int lane = threadIdx.x;
  const int half = lane >> 4;   // 0: lanes 0-15, 1: lanes 16-31
  const int l15 = lane & 15;

  const float* Q = qkv + (size_t)(at * 3 + 0) * BNH + (size_t)b * NN * HH;
  const float* K = qkv + (size_t)(at * 3 + 1) * BNH + (size_t)b * NN * HH;
  const float* V = qkv + (size_t)(at * 3 + 2) * BNH + (size_t)b * NN * HH;
  float* O = abc + (size_t)at * BNH + (size_t)b * NN * HH;

  __shared__ float lds_k[2][16 * 68];  // stride 68: conflict-free stride-4 gathers
  __shared__ float lds_p[16 * 17];
  const unsigned lds_k_base = (unsigned)(uintptr_t)(&lds_k[0][0]);

  const int q0 = tile * 16;

  // A fragments of Q (16 K-chunks of 4). 16x4 f32 A-layout:
  // lanes0-15 hold (M=lane, K=k0,k0+1); lanes16-31 hold (M=lane-16, K=k0+2,k0+3)
  v2f aq[16];
#pragma unroll
  for (int kk = 0; kk < 16; ++kk) {
    int ka = kk * 4 + half * 2;
    aq[kk].x = Q[(q0 + l15) * HH + ka];
    aq[kk].y = Q[(q0 + l15) * HH + ka + 1];
  }

  v8f acc[4];
  float mrow[8], lrow[8];
#pragma unroll
  for (int hc = 0; hc < 4; ++hc)
#pragma unroll
    for (int j = 0; j < 8; ++j) acc[hc][j] = 0.f;
#pragma unroll
  for (int r = 0; r < 8; ++r) {
    mrow[r] = -__builtin_huge_valf();
    lrow[r] = 0.f;
  }

  // TDM prefetch of first K tile
  tdm_load_tile16x64(lds_k_base, K);

  for (int kt = 0; kt < NN / 16; ++kt) {
    const int k0 = kt * 16;
    const int cur = kt & 1;
    // TDM prefetch of next K tile into the other buffer (last iter: harmless
    // reload of tile 0 so the in-loop wait threshold stays constant).
    const int ktn = (kt + 1 < NN / 16) ? (kt + 1) : 0;
    tdm_load_tile16x64(lds_k_base + (cur ^ 1) * (16 * 68 * 4),
                       K + (size_t)ktn * 16 * HH);
    // prefetch next V tile into cache (global_prefetch_b8)
    __builtin_prefetch(V + (size_t)ktn * 16 * HH + lane * 32, 0, 3);
    // current tile resident when TENSORcnt <= 1 (TDM in-order per wave)
    __builtin_amdgcn_s_wait_tensorcnt(1);

    const float* kb = &lds_k[cur][0];

    // S = Q * K^T : B[kappa][n] = K[k0+n][kappa]; two chains for ILP
    v8f s0, s1;
#pragma unroll
    for (int j = 0; j < 8; ++j) {
      s0[j] = 0.f;
      s1[j] = 0.f;
    }
#pragma unroll
    for (int kk = 0; kk < 16; kk += 2) {
      int ka0 = kk * 4 + half * 2;
      int ka1 = (kk + 1) * 4 + half * 2;
      v2f bk0, bk1;
      bk0.x = kb[l15 * 68 + ka0];
      bk0.y = kb[l15 * 68 + ka0 + 1];
      bk1.x = kb[l15 * 68 + ka1];
      bk1.y = kb[l15 * 68 + ka1 + 1];
      s0 = wmma4(aq[kk], bk0, s0);
      s1 = wmma4(aq[kk + 1], bk1, s1);
    }

    // online softmax per query row (rows live per-VGPR, cols across 16 lanes)
#pragma unroll
    for (int r = 0; r < 8; ++r) {
      float sc = (s0[r] + s1[r]) * SCALE_F;
      float tmax = rowmax16(sc);
      float mnew = fmaxf(mrow[r], tmax);
      float corr = __expf(mrow[r] - mnew);
      mrow[r] = mnew;
      float pe = __expf(sc - mnew);
      lrow[r] = lrow[r] * corr + rowsum16(pe);
#pragma unroll
      for (int hc = 0; hc < 4; ++hc) acc[hc][r] *= corr;
      lds_p[(r + half * 8) * 17 + l15] = pe;  // P[query][key]
    }
    __syncthreads();

    // acc += P(16x16) * V(16x64), 4 K-chunks x 4 col-tiles (4 chains)
#pragma unroll
    for (int kk = 0; kk < 4; ++kk) {
      int ka = kk * 4 + half * 2;
      v2f ap;
      ap.x = lds_p[l15 * 17 + ka];
      ap.y = lds_p[l15 * 17 + ka + 1];
#pragma unroll
      for (int hc = 0; hc < 4; ++hc) {
        v2f bv;
        bv.x = V[(k0 + ka) * HH + hc * 16 + l15];
        bv.y = V[(k0 + ka + 1) * HH + hc * 16 + l15];
        acc[hc] = wmma4(ap, bv, acc[hc]);
      }
    }
    __syncthreads();
  }
  __builtin_amdgcn_s_wait_tensorcnt(0);

  // normalize and store (D layout: VGPR r -> rows r / r+8)
#pragma unroll
  for (int r = 0; r < 8; ++r) {
    float inv = 1.f / lrow[r];
    int row = q0 + r + half * 8;
#pragma unroll
    for (int hc = 0; hc < 4; ++hc)
      O[row * HH + hc * 16 + l15] = acc[hc][r] * inv;
  }
}

// ---------------- 3) x = [a,b,c,h_cur](BNx256) @ Wg(256x192), WMMA ----------------
__global__ __launch_bounds__(32) void gemm_wg_kernel(
    const float* __restrict__ abc, const float* __restrict__ h_cur,
    const float* __restrict__ Wg, float* __restrict__ x) {
  int t = blockIdx.x;  // 16-row tiles over B*N; N%16==0 so no b crossing
  int b = t / (NN / 16);
  int n0 = (t % (NN / 16)) * 16;
  int lane = threadIdx.x;
  int half = lane >> 4;
  int l15 = lane & 15;

  const float* S[4];
  S[0] = abc + (size_t)b * NN * HH;
  S[1] = abc + (size_t)BNH + (size_t)b * NN * HH;
  S[2] = abc + (size_t)2 * BNH + (size_t)b * NN * HH;
  S[3] = h_cur + (size_t)b * NN * HH;

  v8f acc[12];
#pragma unroll
  for (int ct = 0; ct < 12; ++ct)
#pragma unroll
    for (int j = 0; j < 8; ++j) acc[ct][j] = 0.f;

  int rowA = n0 + l15;
  for (int kk = 0; kk < 64; ++kk) {
    int ka = kk * 4 + half * 2;        // K index in [0,256)
    const float* Sp = S[ka >> 6];      // same block for whole chunk of 4
    int kr = ka & 63;
    v2f a;
    a.x = Sp[rowA * HH + kr];
    a.y = Sp[rowA * HH + kr + 1];
#pragma unroll
    for (int ct = 0; ct < 12; ++ct) {
      v2f bw;
      bw.x = Wg[ka * HX + ct * 16 + l15];
      bw.y = Wg[(ka + 1) * HX + ct * 16 + l15];
      acc[ct] = wmma4(a, bw, acc[ct]);
    }
  }
#pragma unroll
  for (int ct = 0; ct < 12; ++ct)
#pragma unroll
    for (int r = 0; r < 8; ++r)
      x[((size_t)b * NN + n0 + r + half * 8) * HX + ct * 16 + l15] =
          acc[ct][r];
}

// ---------------- 4) al_s / al_d ----------------
__global__ void al_kernel(const float* __restrict__ x,
                          const float* __restrict__ att_src,
                          const float* __restrict__ att_dst,
                          float* __restrict__ al_s, float* __restrict__ al_d) {
  int i = blockIdx.x * 256 + threadIdx.x;
  if (i >= BN) return;
  const float* xr = x + (size_t)i * HX;
  float s = 0.f, d = 0.f;
  for (int j = 0; j < HX; ++j) {
    float v = xr[j];
    s += v * att_src[j];
    d += v * att_dst[j];
  }
  al_s[i] = s;
  al_d[i] = d;
}

// ---------------- 5) init segment buffers ----------------
__global__ void init_kernel(float* __restrict__ out_agg,
                            unsigned* __restrict__ segmax,
                            float* __restrict__ denom) {
  int i = blockIdx.x * 256 + threadIdx.x;
  if (i < BN * HX) out_agg[i] = 0.f;
  if (i < BN) {
    segmax[i] = ENC_NEG_INF;
    denom[i] = 0.f;
  }
}

// ---------------- 6) edge pass 1: leaky_relu + segment max ----------------
__global__ void edge_max_kernel(const int* __restrict__ eidx,
                                const float* __restrict__ al_s,
                                const float* __restrict__ al_d,
                                float* __restrict__ e_buf,
                                unsigned* __restrict__ segmax) {
  int t = blockIdx.x * 256 + threadIdx.x;
  if (t >= BB * EE) return;
  int b = t / EE, e = t % EE;
  int s = eidx[e], d = eidx[EE + e];
  float ee = al_s[b * NN + s] + al_d[b * NN + d];
  ee = (ee > 0.f) ? ee : ee * NEG_SLOPE_F;
  e_buf[t] = ee;
  atomicMax(&segmax[b * NN + d], fenc(ee));
}

// ---------------- 7) edge pass 2: exp + segment sum ----------------
__global__ void edge_exp_kernel(const int* __restrict__ eidx,
                                float* __restrict__ e_buf,
                                const unsigned* __restrict__ segmax,
                                float* __restrict__ denom) {
  int t = blockIdx.x * 256 + threadIdx.x;
  if (t >= BB * EE) return;
  int b = t / EE, e = t % EE;
  int d = eidx[EE + e];
  float m = fdec(segmax[b * NN + d]);
  float ex = __expf(e_buf[t] - m);
  e_buf[t] = ex;
  atomicAdd(&denom[b * NN + d], ex);
}

// ---------------- 8) edge pass 3: scatter alpha * x[src] (warp per edge) ----------------
__global__ void edge_scatter_kernel(const int* __restrict__ eidx,
                                    const float* __restrict__ e_buf,
                                    const float* __restrict__ denom,
                                    const float* __restrict__ x,
                                    float* __restrict__ out_agg) {
  int w = (blockIdx.x * 256 + threadIdx.x) >> 5;
  int lane = threadIdx.x & 31;
  if (w >= BB * EE) return;
  int b = w / EE, e = w % EE;
  int s = eidx[e], d = eidx[EE + e];
  float alpha = e_buf[w] / (denom[b * NN + d] + 1e-16f);
  const float* xs = x + ((size_t)(b * NN + s)) * HX;
  float* od = out_agg + ((size_t)(b * NN + d)) * HX;
#pragma unroll
  for (int j = lane; j < HX; j += 32) atomicAdd(&od[j], alpha * xs[j]);
}

// ---------------- 9) GRU epilogue ----------------
__global__ void gru_kernel(const float* __restrict__ out_agg,
                           const float* __restrict__ bg,
                           const float* __restrict__ h_cur,
                           float* __restrict__ hout) {
  int i = blockIdx.x * 256 + threadIdx.x;
  if (i >= BNH) return;
  int h = i & 63;
  size_t row = (size_t)(i >> 6);
  const float* o = out_agg + row * HX;
  float ccz = o[64 + h] + bg[64 + h];
  float ccn = o[128 + h] + bg[128 + h];
  float z = 1.f / (1.f + __expf(-ccz));
  float nn = tanhf(ccn);
  hout[i] = (1.f - z) * nn + z * h_cur[i];
}

// ---------------- launcher ----------------
extern "C" void kernel_launch(void* const* d_in, const int* in_sizes, int n_in,
                              void* d_out, int out_size, void* d_ws,
                              size_t ws_size, hipStream_t stream) {
  const float* in_t = (const float*)d_in[0];
  const float* h_cur = (const float*)d_in[1];
  const int* eidx = (const int*)d_in[2];
  const float* Wqa = (const float*)d_in[3];
  const float* bqa = (const float*)d_in[4];
  const float* Wka = (const float*)d_in[5];
  const float* bka = (const float*)d_in[6];
  const float* Wva = (const float*)d_in[7];
  const float* bva = (const float*)d_in[8];
  const float* Wqb = (const float*)d_in[9];
  const float* bqb = (const float*)d_in[10];
  const float* Wkb = (const float*)d_in[11];
  const float* bkb = (const float*)d_in[12];
  const float* Wvb = (const float*)d_in[13];
  const float* bvb = (const float*)d_in[14];
  const float* Wqc = (const float*)d_in[15];
  const float* bqc = (const float*)d_in[16];
  const float* Wkc = (const float*)d_in[17];
  const float* bkc = (const float*)d_in[18];
  const float* Wvc = (const float*)d_in[19];
  const float* bvc = (const float*)d_in[20];
  const float* Wg = (const float*)d_in[21];
  const float* att_src = (const float*)d_in[22];
  const float* att_dst = (const float*)d_in[23];
  const float* bg = (const float*)d_in[24];
  float* hout = (float*)d_out;

  float* ws = (float*)d_ws;
  // phase 1/2 layout
  float* qkv = ws;                    // 9*BNH
  float* abc = ws + (size_t)9 * BNH;  // 3*BNH
  // phase 3 overlay on the (dead) qkv region
  float* x = ws;                      // BN*192
  float* out_agg = ws + 1536000;      // BN*192
  float* e_buf = ws + 3072000;        // B*E
  float* al_s = ws + 3328000;         // BN
  float* al_d = ws + 3336000;         // BN
  unsigned* segmax = (unsigned*)(ws + 3344000);  // BN
  float* denom = ws + 3352000;        // BN

  // 1) projections
  proj_kernel<<<(BNH + 255) / 256, 256, 0, stream>>>(
      in_t, Wqa, bqa, Wka, bka, Wva, bva, Wqb, bqb, Wkb, bkb, Wvb, bvb, Wqc,
      bqc, Wkc, bkc, Wvc, bvc, qkv);
  // 2) three cross attentions (flash, WMMA f32, TDM-fed K tiles)
  attn_kernel<<<dim3(NN / 16, BB, 3), 32, 0, stream>>>(qkv, abc);
  // 3) fused GEMM -> x
  gemm_wg_kernel<<<BB * (NN / 16), 32, 0, stream>>>(abc, h_cur, Wg, x);
  // 4) attention logits per node
  al_kernel<<<(BN + 255) / 256, 256, 0, stream>>>(x, att_src, att_dst, al_s,
                                                  al_d);
  // 5) init segment buffers
  init_kernel<<<(BN * HX + 255) / 256, 256, 0, stream>>>(out_agg, segmax,
                                                         denom);
  // 6-8) GAT edge softmax + aggregation
  edge_max_kernel<<<(BB * EE + 255) / 256, 256, 0, stream>>>(eidx, al_s, al_d,
                                                             e_buf, segmax);
  edge_exp_kernel<<<(BB * EE + 255) / 256, 256, 0, stream>>>(eidx, e_buf,
                                                             segmax, denom);
  edge_scatter_kernel<<<(BB * EE * 32 + 255) / 256, 256, 0, stream>>>(
      eidx, e_buf, denom, x, out_agg);
  // 9) GRU
  gru_kernel<<<(BNH + 255) / 256, 256, 0, stream>>>(out_agg, bg, h_cur, hout);
}